// CDF_origin_35407710388891
// MI455X (gfx1250) — compile-verified
//
#include <hip/hip_runtime.h>
#include <math.h>

#define C_CH 320
#define N_S  65536
#define TPB  256
#define VEC  4
#define ITERS 4
#define CHUNK (TPB * VEC * ITERS)        /* 4096 samples per block */
#define CHUNKS_PER_CH (N_S / CHUNK)      /* 16 */
#define NPARAM 43

typedef float v4f __attribute__((ext_vector_type(4)));

// Per-lane param descriptor: sel | start<<4 | stride<<10
// sel: index into ptab {M0,M1,M2,M3,b0,b1,b2,b3,f0,f1,f2}
// start: first tid of the group; stride: per-channel element stride
__device__ __constant__ unsigned short kDesc[NPARAM] = {
    0x0C00, 0x0C00, 0x0C00,                                                 // M0: c*3
    0x2431, 0x2431, 0x2431, 0x2431, 0x2431, 0x2431, 0x2431, 0x2431, 0x2431, // M1: c*9
    0x24C2, 0x24C2, 0x24C2, 0x24C2, 0x24C2, 0x24C2, 0x24C2, 0x24C2, 0x24C2, // M2: c*9
    0x0D53, 0x0D53, 0x0D53,                                                 // M3: c*3
    0x0D84, 0x0D84, 0x0D84,                                                 // b0
    0x0DB5, 0x0DB5, 0x0DB5,                                                 // b1
    0x0DE6, 0x0DE6, 0x0DE6,                                                 // b2
    0x0617,                                                                 // b3: c*1
    0x0E28, 0x0E28, 0x0E28,                                                 // f0
    0x0E59, 0x0E59, 0x0E59,                                                 // f1
    0x0E8A, 0x0E8A, 0x0E8A,                                                 // f2
};

// ---- hardware tanh: v_tanh_f32 on the gfx1250 trans pipe, guaranteed ----
__device__ __forceinline__ float htanh(float x) {
#if __has_builtin(__builtin_amdgcn_tanhf)
    return __builtin_amdgcn_tanhf(x);
#elif __has_builtin(__builtin_amdgcn_tanh_f32)
    return __builtin_amdgcn_tanh_f32(x);
#else
    float r;
    asm("v_tanh_f32 %0, %1" : "=v"(r) : "v"(x));
    return r;
#endif
}

__device__ __forceinline__ float softplus_f(float x) {
    // stable log(1+exp(x)); executed ~43x per block, accuracy over speed
    return (x > 20.0f) ? x : log1pf(expf(x));
}

struct Params {
    float m0[3], m1[9], m2[9], m3[3];
    float b0[3], b1[3], b2[3], b3;
    float t0[3], t1[3], t2[3];
};

__device__ __forceinline__ float eval_one(float x, const Params& p) {
    // layer 0: 1 -> 3, gated
    float h0 = fmaf(p.m0[0], x, p.b0[0]);
    float h1 = fmaf(p.m0[1], x, p.b0[1]);
    float h2 = fmaf(p.m0[2], x, p.b0[2]);
    h0 = fmaf(p.t0[0], htanh(h0), h0);
    h1 = fmaf(p.t0[1], htanh(h1), h1);
    h2 = fmaf(p.t0[2], htanh(h2), h2);
    // layer 1: 3 -> 3, gated
    float g0 = fmaf(p.m1[0], h0, p.b1[0]); g0 = fmaf(p.m1[1], h1, g0); g0 = fmaf(p.m1[2], h2, g0);
    float g1 = fmaf(p.m1[3], h0, p.b1[1]); g1 = fmaf(p.m1[4], h1, g1); g1 = fmaf(p.m1[5], h2, g1);
    float g2 = fmaf(p.m1[6], h0, p.b1[2]); g2 = fmaf(p.m1[7], h1, g2); g2 = fmaf(p.m1[8], h2, g2);
    g0 = fmaf(p.t1[0], htanh(g0), g0);
    g1 = fmaf(p.t1[1], htanh(g1), g1);
    g2 = fmaf(p.t1[2], htanh(g2), g2);
    // layer 2: 3 -> 3, gated
    h0 = fmaf(p.m2[0], g0, p.b2[0]); h0 = fmaf(p.m2[1], g1, h0); h0 = fmaf(p.m2[2], g2, h0);
    h1 = fmaf(p.m2[3], g0, p.b2[1]); h1 = fmaf(p.m2[4], g1, h1); h1 = fmaf(p.m2[5], g2, h1);
    h2 = fmaf(p.m2[6], g0, p.b2[2]); h2 = fmaf(p.m2[7], g1, h2); h2 = fmaf(p.m2[8], g2, h2);
    h0 = fmaf(p.t2[0], htanh(h0), h0);
    h1 = fmaf(p.t2[1], htanh(h1), h1);
    h2 = fmaf(p.t2[2], htanh(h2), h2);
    // layer 3: 3 -> 1
    float y = fmaf(p.m3[0], h0, p.b3);
    y = fmaf(p.m3[1], h1, y);
    y = fmaf(p.m3[2], h2, y);
    return y;
}

__global__ __launch_bounds__(TPB) void cdf_kernel(
    const float* __restrict__ inputs,
    const float* __restrict__ M0, const float* __restrict__ b0, const float* __restrict__ f0,
    const float* __restrict__ M1, const float* __restrict__ b1, const float* __restrict__ f1,
    const float* __restrict__ M2, const float* __restrict__ b2, const float* __restrict__ f2,
    const float* __restrict__ M3, const float* __restrict__ b3,
    float* __restrict__ out)
{
    __shared__ const float* ptab[12];
    __shared__ float sp[48];   // [0:24) raw M -> softplus; [24:34) biases; [34:43) f -> tanh
    const int tid   = threadIdx.x;
    const int c     = blockIdx.x / CHUNKS_PER_CH;
    const int chunk = blockIdx.x % CHUNKS_PER_CH;

    // lane 0 publishes the 11 base pointers: straight-line ds_store_b128s
    if (tid == 0) {
        ptab[0] = M0; ptab[1] = M1; ptab[2]  = M2; ptab[3] = M3;
        ptab[4] = b0; ptab[5] = b1; ptab[6]  = b2; ptab[7] = b3;
        ptab[8] = f0; ptab[9] = f1; ptab[10] = f2;
    }
    __syncthreads();

    // ---- stage + transform this channel's 43 params; zero data-dependent branches ----
    if (tid < NPARAM) {
        const unsigned d   = kDesc[tid];           // global_load_u16 from constant mem
        const int sel      = d & 0xF;
        const int start    = (d >> 4) & 0x3F;
        const int stride   = (d >> 10) & 0xF;
        const float* src   = ptab[sel] + c * stride + (tid - start);

        // async DMA into LDS (ASYNCcnt path); generic LDS address low 32 bits
        // == byte offset within the workgroup allocation
        unsigned lds_off = (unsigned)(unsigned long long)(&sp[tid]);
        asm volatile("global_load_async_to_lds_b32 %0, %1, off"
                     :: "v"(lds_off), "v"(src) : "memory");
        asm volatile("s_wait_asynccnt 0x0" ::: "memory");

        // own-lane transform (this wave's own wait covers its own lanes' slots)
        float raw    = sp[tid];
        float spv    = softplus_f(raw);
        float thv    = htanh(raw);
        float cooked = (tid < 24) ? spv : ((tid >= 34) ? thv : raw);
        sp[tid] = cooked;
    }
    __syncthreads();

    // ---- hoist params into registers ----
    Params p;
    #pragma unroll
    for (int k = 0; k < 3; ++k) p.m0[k] = sp[k];
    #pragma unroll
    for (int k = 0; k < 9; ++k) p.m1[k] = sp[3 + k];
    #pragma unroll
    for (int k = 0; k < 9; ++k) p.m2[k] = sp[12 + k];
    #pragma unroll
    for (int k = 0; k < 3; ++k) p.m3[k] = sp[21 + k];
    #pragma unroll
    for (int k = 0; k < 3; ++k) p.b0[k] = sp[24 + k];
    #pragma unroll
    for (int k = 0; k < 3; ++k) p.b1[k] = sp[27 + k];
    #pragma unroll
    for (int k = 0; k < 3; ++k) p.b2[k] = sp[30 + k];
    p.b3 = sp[33];
    #pragma unroll
    for (int k = 0; k < 3; ++k) p.t0[k] = sp[34 + k];
    #pragma unroll
    for (int k = 0; k < 3; ++k) p.t1[k] = sp[37 + k];
    #pragma unroll
    for (int k = 0; k < 3; ++k) p.t2[k] = sp[40 + k];

    // ---- stream 4096 samples: b128 nontemporal in/out ----
    const size_t base = (size_t)c * N_S + (size_t)chunk * CHUNK;
    const v4f* __restrict__ in4  = reinterpret_cast<const v4f*>(inputs + base);
    v4f*       __restrict__ out4 = reinterpret_cast<v4f*>(out + base);

    v4f xv[ITERS];
    #pragma unroll
    for (int it = 0; it < ITERS; ++it)
        xv[it] = __builtin_nontemporal_load(in4 + it * TPB + tid);

    #pragma unroll
    for (int it = 0; it < ITERS; ++it) {
        v4f y;
        y.x = eval_one(xv[it].x, p);
        y.y = eval_one(xv[it].y, p);
        y.z = eval_one(xv[it].z, p);
        y.w = eval_one(xv[it].w, p);
        __builtin_nontemporal_store(y, out4 + it * TPB + tid);
    }
}

extern "C" void kernel_launch(void* const* d_in, const int* in_sizes, int n_in,
                              void* d_out, int out_size, void* d_ws, size_t ws_size,
                              hipStream_t stream) {
    (void)in_sizes; (void)n_in; (void)out_size; (void)d_ws; (void)ws_size;
    // setup_inputs order: inputs, M0,b0,f0, M1,b1,f1, M2,b2,f2, M3,b3, stop_gradient
    const float* inputs = (const float*)d_in[0];
    const float* M0 = (const float*)d_in[1];
    const float* b0 = (const float*)d_in[2];
    const float* f0 = (const float*)d_in[3];
    const float* M1 = (const float*)d_in[4];
    const float* b1 = (const float*)d_in[5];
    const float* f1 = (const float*)d_in[6];
    const float* M2 = (const float*)d_in[7];
    const float* b2 = (const float*)d_in[8];
    const float* f2 = (const float*)d_in[9];
    const float* M3 = (const float*)d_in[10];
    const float* b3 = (const float*)d_in[11];
    float* out = (float*)d_out;

    dim3 grid(C_CH * CHUNKS_PER_CH), block(TPB);
    hipLaunchKernelGGL(cdf_kernel, grid, block, 0, stream,
                       inputs, M0, b0, f0, M1, b1, f1, M2, b2, f2, M3, b3, out);
}